// Denoising_ResNet_68719477236
// MI455X (gfx1250) — compile-verified
//
#include <hip/hip_runtime.h>

// CDNA5 / gfx1250 — fp32 WMMA (16x16x4) path for x + conv1x1(box_mean3x3(x)) + b
// Tensor: [B=32, C=64, H=128, W=128] fp32.

typedef __attribute__((ext_vector_type(2))) float v2f;
typedef __attribute__((ext_vector_type(8))) float v8f;

#define B_DIM 32
#define C_CH  64
#define H_DIM 128
#define W_DIM 128
#define WSTRIDE 68    // padded LDS row stride for weights  (bank stride 4 over o)
#define MSTRIDE 136   // padded LDS row stride for row data (2*136 % 64 == 16)

__global__ __launch_bounds__(256)
void denoise_resnet_row_kernel(const float* __restrict__ x,
                               const float* __restrict__ cw,   // [64,64] row-major (o,c)
                               const float* __restrict__ cb,   // [64]
                               float* __restrict__ out)
{
    __shared__ float s_w[C_CH * WSTRIDE];      // weights, padded
    __shared__ float s_bias[C_CH];
    __shared__ float s_vs[C_CH * MSTRIDE];     // vertical clipped sums
    __shared__ float s_mid[C_CH * MSTRIDE];    // middle row of x (residual)
    __shared__ float s_m[C_CH * MSTRIDE];      // box mean (B operand)

    const int t  = threadIdx.x;
    const int bh = blockIdx.x;        // 0 .. B*H-1
    const int b  = bh >> 7;           // / H_DIM
    const int h  = bh & (H_DIM - 1);

    // ---- weights + bias into LDS ----
    for (int i = t; i < C_CH * C_CH; i += 256) {
        const int o = i >> 6;
        const int c = i & 63;
        s_w[o * WSTRIDE + c] = cw[i];
    }
    if (t < C_CH) s_bias[t] = cb[t];

    // ---- vertical clipped 3-row sums (coalesced global loads), keep mid row ----
    for (int i = t; i < C_CH * W_DIM; i += 256) {
        const int c = i >> 7;
        const int w = i & (W_DIM - 1);
        const float* p = x + (((size_t)b * C_CH + c) * H_DIM + h) * W_DIM + w;
        const float mid = p[0];
        float s = mid;
        if (h > 0)         s += p[-W_DIM];
        if (h < H_DIM - 1) s += p[ W_DIM];
        s_mid[c * MSTRIDE + w] = mid;
        s_vs [c * MSTRIDE + w] = s;
    }
    __syncthreads();

    // ---- horizontal clipped sum, divide by true in-bounds count ----
    const float rh = (h == 0 || h == H_DIM - 1) ? 2.0f : 3.0f;
    for (int i = t; i < C_CH * W_DIM; i += 256) {
        const int c = i >> 7;
        const int w = i & (W_DIM - 1);
        float s  = s_vs[c * MSTRIDE + w];
        float rw = 3.0f;
        if (w > 0)         s += s_vs[c * MSTRIDE + w - 1]; else rw -= 1.0f;
        if (w < W_DIM - 1) s += s_vs[c * MSTRIDE + w + 1]; else rw -= 1.0f;
        s_m[c * MSTRIDE + w] = s / (rh * rw);
    }
    __syncthreads();

    // ---- GEMM: D[64 x 128] = W[64x64] * M[64x128] via V_WMMA_F32_16X16X4_F32 ----
    // 8 waves; wave owns 16-column group; K=64 consumed in 16 k-steps of 4;
    // 4 accumulators cover the 4 output-channel tiles.
    const int wave  = t >> 5;          // wave32
    const int lane  = t & 31;
    const int lhi   = lane >> 4;       // 0: lanes 0-15, 1: lanes 16-31
    const int llo   = lane & 15;
    const int nbase = wave * 16;       // column group within the row

    v8f zero = {};
    v8f acc0 = zero, acc1 = zero, acc2 = zero, acc3 = zero;

#pragma unroll
    for (int ks = 0; ks < 16; ++ks) {
        const int k0 = ks * 4 + 2 * lhi;   // K index for .x ; .y uses k0+1
        // B fragment (4x16): row K striped across lanes within a VGPR
        v2f bf;
        bf.x = s_m[(k0    ) * MSTRIDE + nbase + llo];
        bf.y = s_m[(k0 + 1) * MSTRIDE + nbase + llo];
        // A fragments (16x4): lane -> row M, VGPR/half-wave -> K
        v2f af;
        af.x = s_w[( 0 + llo) * WSTRIDE + k0];
        af.y = s_w[( 0 + llo) * WSTRIDE + k0 + 1];
        acc0 = __builtin_amdgcn_wmma_f32_16x16x4_f32(false, af, false, bf,
                                                     (short)0, acc0, false, false);
        af.x = s_w[(16 + llo) * WSTRIDE + k0];
        af.y = s_w[(16 + llo) * WSTRIDE + k0 + 1];
        acc1 = __builtin_amdgcn_wmma_f32_16x16x4_f32(false, af, false, bf,
                                                     (short)0, acc1, false, false);
        af.x = s_w[(32 + llo) * WSTRIDE + k0];
        af.y = s_w[(32 + llo) * WSTRIDE + k0 + 1];
        acc2 = __builtin_amdgcn_wmma_f32_16x16x4_f32(false, af, false, bf,
                                                     (short)0, acc2, false, false);
        af.x = s_w[(48 + llo) * WSTRIDE + k0];
        af.y = s_w[(48 + llo) * WSTRIDE + k0 + 1];
        acc3 = __builtin_amdgcn_wmma_f32_16x16x4_f32(false, af, false, bf,
                                                     (short)0, acc3, false, false);
    }

    // ---- epilogue: out = x_mid + D + bias ----
    // C/D layout: VGPR r holds rows M = r (lanes 0-15) and M = r+8 (lanes 16-31).
    float* ob = out + (((size_t)b * C_CH) * H_DIM + h) * W_DIM;
    const int wg = nbase + llo;                 // global column within the row
    v8f accs[4] = {acc0, acc1, acc2, acc3};
#pragma unroll
    for (int ot = 0; ot < 4; ++ot) {
#pragma unroll
        for (int r = 0; r < 8; ++r) {
            const int o = ot * 16 + r + 8 * lhi;
            const float v = accs[ot][r] + s_bias[o] + s_mid[o * MSTRIDE + wg];
            ob[(size_t)o * (H_DIM * W_DIM) + wg] = v;
        }
    }
}

extern "C" void kernel_launch(void* const* d_in, const int* in_sizes, int n_in,
                              void* d_out, int out_size, void* d_ws, size_t ws_size,
                              hipStream_t stream) {
    (void)in_sizes; (void)n_in; (void)d_ws; (void)ws_size; (void)out_size;
    const float* x  = (const float*)d_in[0];   // [32,64,128,128]
    const float* cw = (const float*)d_in[1];   // [64,64]
    const float* cb = (const float*)d_in[2];   // [64]
    float* out = (float*)d_out;                // [32,64,128,128]

    dim3 grid(B_DIM * H_DIM);                  // one workgroup per (b, h) row
    dim3 block(256);                           // 8 wave32 waves
    denoise_resnet_row_kernel<<<grid, block, 0, stream>>>(x, cw, cb, out);
}